// SAGE_OneHot2_42150809043596
// MI455X (gfx1250) — compile-verified
//
#include <hip/hip_runtime.h>
#include <hip/hip_bf16.h>

typedef __attribute__((ext_vector_type(2))) float v2f;
typedef __attribute__((ext_vector_type(8))) float v8f;

#define NODES_PER_BLK 48    // 3 M-tiles of 16 per wave -> 3x weight reuse
#define MT 3                // M-tiles per wave
#define KDIM 256            // stacked K: [mean(128) | x(128)]
#define A_STRIDE 260        // padded row: banks (4m+k)%64 -> conflict-free frag reads

// ---------------- zero accumulators ----------------
__global__ void sage_zero_kernel(float* __restrict__ sum, float* __restrict__ deg,
                                 long long n_sum, long long n_deg) {
    long long i = (long long)blockIdx.x * blockDim.x + threadIdx.x;
    long long stride = (long long)gridDim.x * blockDim.x;
    for (long long k = i; k < n_sum; k += stride) sum[k] = 0.0f;
    for (long long k = i; k < n_deg; k += stride) deg[k] = 0.0f;
}

// ---------------- pack [W_l ; W_r] into one contiguous 256x128 matrix ----------------
__global__ void sage_packw_kernel(const float* __restrict__ Wl1, const float* __restrict__ Wr1,
                                  const float* __restrict__ Wl2, const float* __restrict__ Wr2,
                                  float* __restrict__ Ws1, float* __restrict__ Ws2) {
    int i = blockIdx.x * blockDim.x + threadIdx.x;   // 0 .. 128*128-1
    if (i < 128 * 128) {
        Ws1[i]             = Wl1[i];
        Ws1[128 * 128 + i] = Wr1[i];
        Ws2[i]             = Wl2[i];
        Ws2[128 * 128 + i] = Wr2[i];
    }
}

// ---------------- edge scatter: one wave32 per edge ----------------
__global__ void sage_scatter_kernel(const float* __restrict__ feat,
                                    const int* __restrict__ src,
                                    const int* __restrict__ dst,
                                    float* __restrict__ sum,
                                    float* __restrict__ deg, int E) {
    int edge = blockIdx.x * 8 + (threadIdx.x >> 5);   // 8 waves per block
    if (edge >= E) return;
    int lane = threadIdx.x & 31;
    int s = src[edge];
    int d = dst[edge];
    // 32 lanes x float4 = 128 channels; rows are 512B, L2-resident
    const float4 v = ((const float4*)(feat + (size_t)s * 128))[lane];
    float* o = sum + (size_t)d * 128 + lane * 4;
    atomicAdd(o + 0, v.x);
    atomicAdd(o + 1, v.y);
    atomicAdd(o + 2, v.z);
    atomicAdd(o + 3, v.w);
    if (lane == 0) atomicAdd(deg + d, 1.0f);
}

// ---------------- fused mean + dual-GEMM via fp32 WMMA ----------------
// out[48 x 128] = [mean | x](48x256) @ Wstack(256x128) + b   (+optional relu)
// 8 waves: wave w owns out-columns [16w,16w+16); each wave does 3 M-tiles,
// sharing each B fragment across 3 independent WMMA chains.
__global__ __launch_bounds__(256)
void sage_gemm_kernel(const float* __restrict__ x,
                      const float* __restrict__ sum,
                      const float* __restrict__ deg,
                      const float* __restrict__ Wst,   // 256x128 row-major [W_l;W_r]
                      const float* __restrict__ bias,
                      float* __restrict__ out,
                      int Nn, int do_relu) {
    __shared__ float As[NODES_PER_BLK][A_STRIDE];   // ~50 KB
    __shared__ float sInv[NODES_PER_BLK];

    const int tid = threadIdx.x;
    const int node_base = blockIdx.x * NODES_PER_BLK;

    if (tid < NODES_PER_BLK) {
        int node = node_base + tid;
        float dg = (node < Nn) ? deg[node] : 1.0f;
        sInv[tid] = 1.0f / fmaxf(dg, 1.0f);
    }
    __syncthreads();

    // Stage A = [mean | x] into LDS. 48*256 = 12288 elems, 256 threads, 48 each.
    for (int idx = tid; idx < NODES_PER_BLK * KDIM; idx += 256) {
        int m = idx >> 8;            // row 0..47
        int c = idx & 255;           // col 0..255
        int node = node_base + m;
        if (node >= Nn) node = Nn - 1;
        float v;
        if (c < 128) v = sum[(size_t)node * 128 + c] * sInv[m];
        else         v = x[(size_t)node * 128 + (c - 128)];
        As[m][c] = v;
    }
    __syncthreads();

    const int lane  = tid & 31;
    const int wave  = tid >> 5;          // 0..7 -> out-column tile
    const int mcol  = lane & 15;         // row within M-tile (A) / col (B)
    const int khalf = lane >> 4;         // selects K pair within step of 4
    const int n     = wave * 16 + mcol;  // output channel 0..127

    // B fragment base: element (k0 + 2*khalf)*128 + n  ->  bq + k0*128
    const float* bq = Wst + n + khalf * 256;
    // A fragment bases: As[t*16+mcol][k0 + 2*khalf]    ->  aq[t] + k0
    const float* aq0 = &As[0 * 16 + mcol][2 * khalf];
    const float* aq1 = &As[1 * 16 + mcol][2 * khalf];
    const float* aq2 = &As[2 * 16 + mcol][2 * khalf];

    v8f acc0 = {}, acc1 = {}, acc2 = {};

    #pragma unroll
    for (int k0 = 0; k0 < KDIM; k0 += 4) {
        v2f b;
        b.x = bq[k0 * 128];
        b.y = bq[k0 * 128 + 128];
        v2f a0 = *(const v2f*)(aq0 + k0);   // ds_load_b64, const offset
        v2f a1 = *(const v2f*)(aq1 + k0);
        v2f a2 = *(const v2f*)(aq2 + k0);
        acc0 = __builtin_amdgcn_wmma_f32_16x16x4_f32(
            false, a0, false, b, (short)0, acc0, false, false);
        acc1 = __builtin_amdgcn_wmma_f32_16x16x4_f32(
            false, a1, false, b, (short)0, acc1, false, false);
        acc2 = __builtin_amdgcn_wmma_f32_16x16x4_f32(
            false, a2, false, b, (short)0, acc2, false, false);
    }

    const float bn = bias[n];
    v8f accs[MT] = {acc0, acc1, acc2};
    #pragma unroll
    for (int t = 0; t < MT; ++t) {
        #pragma unroll
        for (int r = 0; r < 8; ++r) {
            int m = t * 16 + r + 8 * khalf;  // C layout: VGPR r -> M=r / M=r+8
            int node = node_base + m;
            if (node < Nn) {
                float v = accs[t][r] + bn;
                if (do_relu) v = fmaxf(v, 0.0f);
                out[(size_t)node * 128 + n] = v;
            }
        }
    }
}

extern "C" void kernel_launch(void* const* d_in, const int* in_sizes, int n_in,
                              void* d_out, int out_size, void* d_ws, size_t ws_size,
                              hipStream_t stream) {
    const float* x    = (const float*)d_in[0];
    const int*   edge = (const int*)d_in[1];     // [2, E] int32
    const float* W_l1 = (const float*)d_in[2];
    const float* W_r1 = (const float*)d_in[3];
    const float* b1   = (const float*)d_in[4];
    const float* W_l2 = (const float*)d_in[5];
    const float* W_r2 = (const float*)d_in[6];
    const float* b2   = (const float*)d_in[7];

    const int N = in_sizes[0] / 128;
    const int E = in_sizes[1] / 2;
    const int* src = edge;
    const int* dst = edge + E;

    float* sum = (float*)d_ws;                       // N*128 f32
    float* h   = sum + (size_t)N * 128;              // N*128 f32
    float* deg = h + (size_t)N * 128;                // N f32
    float* Ws1 = deg + N;                            // 256*128 f32 stacked weights L1
    float* Ws2 = Ws1 + 256 * 128;                    // 256*128 f32 stacked weights L2

    float* outp = (float*)d_out;

    const long long nSum = (long long)N * 128;
    const int zeroBlocks = 2048;
    const int scatBlocks = (E + 7) / 8;              // one wave32 per edge
    const int gemmBlocks = (N + NODES_PER_BLK - 1) / NODES_PER_BLK;
    const int packBlocks = (128 * 128 + 255) / 256;

    sage_packw_kernel<<<packBlocks, 256, 0, stream>>>(W_l1, W_r1, W_l2, W_r2, Ws1, Ws2);

    // ---- layer 1 ----
    sage_zero_kernel<<<zeroBlocks, 256, 0, stream>>>(sum, deg, nSum, N);
    sage_scatter_kernel<<<scatBlocks, 256, 0, stream>>>(x, src, dst, sum, deg, E);
    sage_gemm_kernel<<<gemmBlocks, 256, 0, stream>>>(x, sum, deg, Ws1, b1, h, N, 1);
    // ---- layer 2 ----
    sage_zero_kernel<<<zeroBlocks, 256, 0, stream>>>(sum, deg, nSum, N);
    sage_scatter_kernel<<<scatBlocks, 256, 0, stream>>>(h, src, dst, sum, deg, E);
    sage_gemm_kernel<<<gemmBlocks, 256, 0, stream>>>(h, sum, deg, Ws2, b2, outp, N, 0);
}